// DMDMachine_78108275245569
// MI455X (gfx1250) — compile-verified
//
#include <hip/hip_runtime.h>
#include <math.h>

// Problem constants
#define BB   512
#define NF   512
#define TT   64
#define HID  1024
#define LAT  2
#define TM   63            // T-1

typedef float v2f __attribute__((ext_vector_type(2)));
typedef float v8f __attribute__((ext_vector_type(8)));

// ---------------------------------------------------------------------------
// Encoder: h = tanh(x·We1 + be1) held in WMMA accumulators only (never stored),
// immediately contracted with We2 -> y[b,l,t] (+be2). One workgroup per
// (b, t-tile of 16). 8 waves; each wave owns 8 h-tiles, register-blocked so
// the B operand (x tile) is loaded once per k-step and reused by 8 WMMAs.
// ---------------------------------------------------------------------------
__global__ __launch_bounds__(256) void k_enc(
    const float* __restrict__ x,   const float* __restrict__ We1,
    const float* __restrict__ be1, const float* __restrict__ We2,
    const float* __restrict__ be2, float* __restrict__ y)
{
    const int b    = blockIdx.y;
    const int t0   = blockIdx.x * 16;
    const int tid  = threadIdx.x;
    const int lane = tid & 31;
    const int wave = tid >> 5;
    const int mrow = lane & 15;          // tile row (A) / tile col (B,C)
    const int khalf = (lane >> 4) * 2;   // k sub-offset 0 or 2
    const int chalf = (lane >> 4) * 8;   // C row offset 0 or 8

    __shared__ float sy[2 * 16];
    if (tid < 32) sy[tid] = 0.0f;
    __syncthreads();

    const float* xb = x + (size_t)b * NF * TT;
    const int hbase = wave * 16;         // tile i covers h0 = hbase + i*128

    v8f acc[8];
    #pragma unroll
    for (int i = 0; i < 8; ++i) acc[i] = (v8f){0.f,0.f,0.f,0.f,0.f,0.f,0.f,0.f};

    #pragma unroll 2
    for (int k0 = 0; k0 < NF; k0 += 4) {
        v2f bmat;                         // B[k=n, n=t] = x[b, n, t]
        bmat.x = xb[(size_t)(k0 + khalf)     * TT + t0 + mrow];
        bmat.y = xb[(size_t)(k0 + khalf + 1) * TT + t0 + mrow];
        const float* A0 = We1 + (size_t)(k0 + khalf)     * HID + hbase + mrow;
        const float* A1 = We1 + (size_t)(k0 + khalf + 1) * HID + hbase + mrow;
        #pragma unroll
        for (int i = 0; i < 8; ++i) {
            v2f a;                        // A[m=h, k=n] = We1[n, h]
            a.x = A0[i * 128];
            a.y = A1[i * 128];
            acc[i] = __builtin_amdgcn_wmma_f32_16x16x4_f32(
                         false, a, false, bmat, (short)0, acc[i], false, false);
        }
    }

    float p0 = 0.0f, p1 = 0.0f;
    #pragma unroll
    for (int i = 0; i < 8; ++i) {
        const int h0 = hbase + i * 128;
        #pragma unroll
        for (int r = 0; r < 8; ++r) {
            const int hh = h0 + r + chalf;
            const float hv = tanhf(acc[i][r] + be1[hh]);
            p0 += hv * We2[hh * LAT + 0];
            p1 += hv * We2[hh * LAT + 1];
        }
    }
    atomicAdd(&sy[0 * 16 + mrow], p0);
    atomicAdd(&sy[1 * 16 + mrow], p1);
    __syncthreads();
    if (tid < 32) {
        const int l = tid >> 4, tt = tid & 15;
        y[(size_t)b * (LAT * TT) + l * TT + t0 + tt] = sy[l * 16 + tt] + be2[l];
    }
}

// ---------------------------------------------------------------------------
// Decoder: hd = tanh(y·Wd1 + bd1) staged in LDS (1024 x 16 f32 = 64 KB),
// then x_ae = hd·Wd2 + bd2 via WMMA (B operand from LDS). One workgroup per
// (b, t-tile); 8 waves x 4 n-tiles each, register-blocked so the 2 ds_loads
// per k-step feed 4 WMMAs.
// ---------------------------------------------------------------------------
__global__ __launch_bounds__(256) void k_dec(
    const float* __restrict__ y,   const float* __restrict__ Wd1,
    const float* __restrict__ bd1, const float* __restrict__ Wd2,
    const float* __restrict__ bd2, float* __restrict__ xae)
{
    const int b    = blockIdx.y;
    const int t0   = blockIdx.x * 16;
    const int tid  = threadIdx.x;
    const int lane = tid & 31;
    const int wave = tid >> 5;
    const int mrow = lane & 15;
    const int khalf = (lane >> 4) * 2;
    const int chalf = (lane >> 4) * 8;

    __shared__ float hd[HID * 16];   // [h][t], 64 KB

    {
        const int tt = tid & 15;
        const int hg = tid >> 4;     // 16 groups over h
        const float y0v = y[(size_t)b * (LAT * TT) + 0 * TT + t0 + tt];
        const float y1v = y[(size_t)b * (LAT * TT) + 1 * TT + t0 + tt];
        for (int h = hg; h < HID; h += 16)
            hd[h * 16 + tt] = tanhf(y0v * Wd1[h] + y1v * Wd1[HID + h] + bd1[h]);
    }
    __syncthreads();

    const int nbase = wave * 16;     // tile i covers n0 = nbase + i*128

    v8f acc[4];
    #pragma unroll
    for (int i = 0; i < 4; ++i) acc[i] = (v8f){0.f,0.f,0.f,0.f,0.f,0.f,0.f,0.f};

    #pragma unroll 2
    for (int k0 = 0; k0 < HID; k0 += 4) {
        v2f bmat;                     // B[k=h, n=t] = hd[h][t]  (from LDS)
        bmat.x = hd[(k0 + khalf)     * 16 + mrow];
        bmat.y = hd[(k0 + khalf + 1) * 16 + mrow];
        const float* A0 = Wd2 + (size_t)(k0 + khalf)     * NF + nbase + mrow;
        const float* A1 = Wd2 + (size_t)(k0 + khalf + 1) * NF + nbase + mrow;
        #pragma unroll
        for (int i = 0; i < 4; ++i) {
            v2f a;                    // A[m=n, k=h] = Wd2[h, n]
            a.x = A0[i * 128];
            a.y = A1[i * 128];
            acc[i] = __builtin_amdgcn_wmma_f32_16x16x4_f32(
                         false, a, false, bmat, (short)0, acc[i], false, false);
        }
    }

    #pragma unroll
    for (int i = 0; i < 4; ++i) {
        const int n0 = nbase + i * 128;
        #pragma unroll
        for (int r = 0; r < 8; ++r) {
            const int n = n0 + r + chalf;
            xae[(size_t)b * (NF * TT) + (size_t)n * TT + t0 + mrow] = acc[i][r] + bd2[n];
        }
    }
}

// ---------------------------------------------------------------------------
// Small kernels
// ---------------------------------------------------------------------------
__global__ void k_init(float* dmd, float* ael, float* prl) {
    if (threadIdx.x == 0 && blockIdx.x == 0) { *dmd = 0.f; *ael = 0.f; *prl = 0.f; }
}

__global__ void k_ae_loss(const float* __restrict__ x,
                          const float* __restrict__ xae, float* out) {
    const int i = blockIdx.x * 256 + threadIdx.x;    // over (b*NF + n)
    float d = x[(size_t)i * TT] - xae[(size_t)i * TT];
    float v = d * d * (1.0f / (float)(BB * NF));
    __shared__ float red[256];
    red[threadIdx.x] = v; __syncthreads();
    for (int w = 128; w > 0; w >>= 1) {
        if (threadIdx.x < w) red[threadIdx.x] += red[threadIdx.x + w];
        __syncthreads();
    }
    if (threadIdx.x == 0) atomicAdd(out, red[0]);
}

// G = y_minus^T y_minus  (63x63, stride 64)
__global__ void k_gram(const float* __restrict__ y, float* __restrict__ G) {
    const int idx = blockIdx.x * 256 + threadIdx.x;
    if (idx >= TM * TM) return;
    const int i = idx / TM, j = idx % TM;
    float s = 0.f;
    for (int r = 0; r < BB * LAT; ++r)
        s += y[(size_t)r * TT + i] * y[(size_t)r * TT + j];
    G[i * 64 + j] = s;
}

// Gauss-Jordan inverse of 63x63 SPD Gram matrix, single workgroup in LDS.
__global__ __launch_bounds__(256) void k_inv(const float* __restrict__ G,
                                             float* __restrict__ Gi) {
    __shared__ float sA[TM][64];
    __shared__ float sI[TM][64];
    __shared__ float fcol[TM];
    const int tid = threadIdx.x;
    for (int idx = tid; idx < TM * 64; idx += 256) {
        const int r = idx >> 6, j = idx & 63;
        sA[r][j] = (j < TM) ? G[r * 64 + j] : 0.f;
        sI[r][j] = (r == j) ? 1.f : 0.f;
    }
    __syncthreads();
    for (int p = 0; p < TM; ++p) {
        const float invp = 1.0f / sA[p][p];
        __syncthreads();
        if (tid < 64)        sA[p][tid] *= invp;
        else if (tid < 128)  sI[p][tid - 64] *= invp;
        __syncthreads();
        if (tid < TM) fcol[tid] = sA[tid][p];
        __syncthreads();
        for (int idx = tid; idx < TM * 64; idx += 256) {
            const int r = idx >> 6, j = idx & 63;
            if (r != p) {
                const float f = fcol[r];
                sA[r][j] -= f * sA[p][j];
                sI[r][j] -= f * sI[p][j];
            }
        }
        __syncthreads();
    }
    for (int idx = tid; idx < TM * 64; idx += 256) {
        const int r = idx >> 6, j = idx & 63;
        Gi[r * 64 + j] = sI[r][j];
    }
}

// P = Ginv · y_minus^T   (63 x 1024)
__global__ void k_pinv(const float* __restrict__ y, const float* __restrict__ Gi,
                       float* __restrict__ P) {
    const int i = blockIdx.y;
    const int r = blockIdx.x * 256 + threadIdx.x;
    float s = 0.f;
    for (int j = 0; j < TM; ++j)
        s += Gi[i * 64 + j] * y[(size_t)r * TT + j];
    P[(size_t)i * (BB * LAT) + r] = s;
}

// Amat = y_plus · P   (1024 x 1024)
__global__ void k_amat(const float* __restrict__ y, const float* __restrict__ P,
                       float* __restrict__ A) {
    const int r = blockIdx.y;
    const int c = blockIdx.x * 256 + threadIdx.x;
    float s = 0.f;
    for (int i = 0; i < TM; ++i)
        s += y[(size_t)r * TT + 1 + i] * P[(size_t)i * (BB * LAT) + c];
    A[(size_t)r * (BB * LAT) + c] = s;
}

// Q = P · y_minus  (63x63, ~= I); proj = I - Q
__global__ void k_q(const float* __restrict__ y, const float* __restrict__ P,
                    float* __restrict__ Q) {
    const int idx = blockIdx.x * 256 + threadIdx.x;
    if (idx >= TM * TM) return;
    const int i = idx / TM, j = idx % TM;
    float s = 0.f;
    for (int r = 0; r < BB * LAT; ++r)
        s += P[(size_t)i * (BB * LAT) + r] * y[(size_t)r * TT + j];
    Q[i * 64 + j] = s;
}

// dmd_loss = || y_plus (I - Q) ||_F^2
__global__ __launch_bounds__(64) void k_dmdloss(const float* __restrict__ y,
                                                const float* __restrict__ Q,
                                                float* out) {
    const int r = blockIdx.y;
    const int j = threadIdx.x;
    float v = 0.f;
    if (j < TM) {
        float s = y[(size_t)r * TT + 1 + j];
        for (int i = 0; i < TM; ++i)
            s -= y[(size_t)r * TT + 1 + i] * Q[i * 64 + j];
        v = s * s;
    }
    __shared__ float red[64];
    red[threadIdx.x] = v; __syncthreads();
    for (int w = 32; w > 0; w >>= 1) {
        if (threadIdx.x < w) red[threadIdx.x] += red[threadIdx.x + w];
        __syncthreads();
    }
    if (threadIdx.x == 0) atomicAdd(out, red[0]);
}

// One matvec step: v_out = Amat · v_in ; also writes y_pred column t.
__global__ __launch_bounds__(256) void k_matvec(const float* __restrict__ A,
                                                const float* __restrict__ vin,
                                                int vstride, float* __restrict__ vout,
                                                float* __restrict__ ypred, int t) {
    const int r = blockIdx.x;
    const int tid = threadIdx.x;
    float s = 0.f;
    for (int c = tid; c < BB * LAT; c += 256)
        s += A[(size_t)r * (BB * LAT) + c] * vin[(size_t)c * vstride];
    __shared__ float red[256];
    red[tid] = s; __syncthreads();
    for (int w = 128; w > 0; w >>= 1) {
        if (tid < w) red[tid] += red[tid + w];
        __syncthreads();
    }
    if (tid == 0) {
        const float v = red[0];
        vout[r] = v;
        ypred[(size_t)r * TT + t] = v;
    }
}

__global__ void k_predloss(const float* __restrict__ y,
                           const float* __restrict__ yp, float* out) {
    const int i = blockIdx.x * 256 + threadIdx.x;     // 65536 total
    const float d = yp[i] - y[i];
    float v = d * d * (1.0f / (float)(BB * LAT * TT));
    __shared__ float red[256];
    red[threadIdx.x] = v; __syncthreads();
    for (int w = 128; w > 0; w >>= 1) {
        if (threadIdx.x < w) red[threadIdx.x] += red[threadIdx.x + w];
        __syncthreads();
    }
    if (threadIdx.x == 0) atomicAdd(out, red[0]);
}

// ---------------------------------------------------------------------------
extern "C" void kernel_launch(void* const* d_in, const int* in_sizes, int n_in,
                              void* d_out, int out_size, void* d_ws, size_t ws_size,
                              hipStream_t stream) {
    (void)in_sizes; (void)n_in; (void)out_size; (void)ws_size;
    const float* x   = (const float*)d_in[0];
    const float* We1 = (const float*)d_in[1];
    const float* be1 = (const float*)d_in[2];
    const float* We2 = (const float*)d_in[3];
    const float* be2 = (const float*)d_in[4];
    const float* Wd1 = (const float*)d_in[5];
    const float* bd1 = (const float*)d_in[6];
    const float* Wd2 = (const float*)d_in[7];
    const float* bd2 = (const float*)d_in[8];

    float* out  = (float*)d_out;
    float* xae  = out;                                  // 16,777,216
    float* yo   = out + (size_t)BB * NF * TT;           //     65,536
    float* dmd  = yo  + (size_t)BB * LAT * TT;          //          1
    float* ael  = dmd + 1;                              //          1
    float* ypr  = ael + 1;                              //     65,536
    float* ama  = ypr + (size_t)BB * LAT * TT;          //  1,048,576
    float* prl  = ama + (size_t)(BB * LAT) * (BB * LAT);//          1

    float* ws = (float*)d_ws;
    float* G  = ws;                 // 64*64
    float* Gi = G  + 4096;          // 64*64
    float* P  = Gi + 4096;          // 63*1024
    float* Q  = P  + TM * BB * LAT; // 64*64
    float* va = Q  + 4096;          // 1024
    float* vb = va + BB * LAT;      // 1024

    k_init<<<1, 32, 0, stream>>>(dmd, ael, prl);

    k_enc<<<dim3(TT / 16, BB), 256, 0, stream>>>(x, We1, be1, We2, be2, yo);
    k_dec<<<dim3(TT / 16, BB), 256, 0, stream>>>(yo, Wd1, bd1, Wd2, bd2, xae);

    k_ae_loss<<<(BB * NF) / 256, 256, 0, stream>>>(x, xae, ael);

    k_gram<<<(TM * TM + 255) / 256, 256, 0, stream>>>(yo, G);
    k_inv<<<1, 256, 0, stream>>>(G, Gi);
    k_pinv<<<dim3((BB * LAT) / 256, TM), 256, 0, stream>>>(yo, Gi, P);
    k_amat<<<dim3((BB * LAT) / 256, BB * LAT), 256, 0, stream>>>(yo, P, ama);
    k_q<<<(TM * TM + 255) / 256, 256, 0, stream>>>(yo, P, Q);
    k_dmdloss<<<dim3(1, BB * LAT), 64, 0, stream>>>(yo, Q, dmd);

    // y_pred[:, t] = Amat^(t+1) y0 via chained matvecs (v0 = yw[:,0])
    const float* vin = yo; int vstride = TT;
    for (int t = 0; t < TT; ++t) {
        float* vout = (t & 1) ? vb : va;
        k_matvec<<<BB * LAT, 256, 0, stream>>>(ama, vin, vstride, vout, ypr, t);
        vin = vout; vstride = 1;
    }

    k_predloss<<<(BB * LAT * TT) / 256, 256, 0, stream>>>(yo, ypr, prl);
}